// L21_Norm_34316788695050
// MI455X (gfx1250) — compile-verified
//
#include <hip/hip_runtime.h>
#include <math.h>

typedef __attribute__((ext_vector_type(2))) float v2f;
typedef __attribute__((ext_vector_type(8))) float v8f;

// Problem dims (fixed by the reference).
static constexpr int NCOLS = 16384;
static constexpr int NROWS = 8192;
static constexpr int ROWB  = 16;                    // row-blocks (grid.y)
static constexpr int ROWS_PER_BLOCK = NROWS / ROWB; // 512

// ---------------------------------------------------------------------------
// Kernel 1: per-column sum of squares over a row slab, reduction done by
// V_WMMA_F32_16X16X4_F32 with an all-ones A matrix:
//   D[m,n] = sum_k 1 * B[k,n] + C[m,n]
// Each wave owns 16 consecutive columns; B is a 4x16 tile of squared inputs.
// Since A is all ones, any bijective placement of the 4 rows onto the 4 K
// slots of a column is correct; only N = lane%16 must match the C/D layout.
// ---------------------------------------------------------------------------
__global__ __launch_bounds__(256) void l21_colsq_wmma(const float* __restrict__ S,
                                                      float* __restrict__ partial) {
    const int lane = threadIdx.x & 31;
    const int wave = threadIdx.x >> 5;
    const int col  = blockIdx.x * 128 + wave * 16 + (lane & 15);
    // Half-wave row offset: lanes 0-15 cover rows r+0 / r+1, lanes 16-31
    // cover rows r+2 / r+3 (exact K permutation is irrelevant: A == ones).
    const int rowOff = (lane < 16) ? 0 : 2;

    const float* p = S + (size_t)(blockIdx.y * ROWS_PER_BLOCK + rowOff) * NCOLS + col;

    v8f c = {0.f, 0.f, 0.f, 0.f, 0.f, 0.f, 0.f, 0.f};
    const v2f ones = {1.0f, 1.0f};

#pragma unroll 4
    for (int it = 0; it < ROWS_PER_BLOCK / 4; ++it) {
        float s0 = p[0];          // row r + rowOff
        float s1 = p[NCOLS];      // row r + rowOff + 1
        v2f b;
        b.x = s0 * s0;
        b.y = s1 * s1;
        // D = ones(16x4) x Bsq(4x16) + C  -> per-column partial sums in every row of D
        c = __builtin_amdgcn_wmma_f32_16x16x4_f32(false, ones, false, b,
                                                  (short)0, c, false, false);
        p += (size_t)4 * NCOLS;
    }

    // D row M=0 lives in VGPR0 of lanes 0-15 (N = lane). All rows are equal.
    if (lane < 16) {
        partial[(size_t)blockIdx.y * NCOLS + col] = c[0];
    }
}

// ---------------------------------------------------------------------------
// Kernel 2: combine ROWB partials per column, sqrt, block-reduce (fixed order).
// 64 blocks x 256 threads == 16384 columns.
// ---------------------------------------------------------------------------
__global__ __launch_bounds__(256) void l21_colreduce(const float* __restrict__ partial,
                                                     float* __restrict__ blockSums) {
    const int col = blockIdx.x * 256 + threadIdx.x;
    float s = 0.0f;
#pragma unroll
    for (int r = 0; r < ROWB; ++r) {
        s += partial[(size_t)r * NCOLS + col];
    }
    float v = sqrtf(s);

    __shared__ float red[256];
    red[threadIdx.x] = v;
    __syncthreads();
#pragma unroll
    for (int off = 128; off > 0; off >>= 1) {
        if (threadIdx.x < off) red[threadIdx.x] += red[threadIdx.x + off];
        __syncthreads();
    }
    if (threadIdx.x == 0) blockSums[blockIdx.x] = red[0];
}

// ---------------------------------------------------------------------------
// Kernel 3: final 64 -> 1 reduction (single block, deterministic tree).
// ---------------------------------------------------------------------------
__global__ __launch_bounds__(64) void l21_final(const float* __restrict__ blockSums,
                                                float* __restrict__ out) {
    __shared__ float red[64];
    red[threadIdx.x] = blockSums[threadIdx.x];
    __syncthreads();
#pragma unroll
    for (int off = 32; off > 0; off >>= 1) {
        if (threadIdx.x < off) red[threadIdx.x] += red[threadIdx.x + off];
        __syncthreads();
    }
    if (threadIdx.x == 0) out[0] = red[0];
}

extern "C" void kernel_launch(void* const* d_in, const int* in_sizes, int n_in,
                              void* d_out, int out_size, void* d_ws, size_t ws_size,
                              hipStream_t stream) {
    (void)in_sizes; (void)n_in; (void)out_size; (void)ws_size;

    const float* S = (const float*)d_in[0];
    float* out = (float*)d_out;

    // Workspace layout: [ ROWB * NCOLS partials | 64 block sums ]
    float* partial   = (float*)d_ws;                    // 16*16384 floats = 1 MiB
    float* blockSums = partial + (size_t)ROWB * NCOLS;  // 64 floats

    dim3 g1(NCOLS / 128, ROWB);  // (128, 16) = 2048 workgroups
    dim3 b1(256);
    l21_colsq_wmma<<<g1, b1, 0, stream>>>(S, partial);

    l21_colreduce<<<NCOLS / 256, 256, 0, stream>>>(partial, blockSums);

    l21_final<<<1, 64, 0, stream>>>(blockSums, out);
}